// Radon_82282983457006
// MI455X (gfx1250) — compile-verified
//
#include <hip/hip_runtime.h>
#include <hip/hip_bf16.h>

// Radon transform, parallel beam, circle=False.
//   x:   [4, 1, 256, 256] f32
//   out: [4, 1, 363, 180] f32   (G = ceil(sqrt(2)*256) = 363, pad_before = 53)
//
// Sampling math (exact algebra of the reference, in unpadded image coords):
//   u = w - 181, v = h - 181
//   ix = c*u + s*v + 181 - 53 = c*u + s*v + 128
//   iy = -s*u + c*v + 128
// Bilinear taps outside [0,255]^2 contribute 0 (pad region is zero and
// out-of-grid taps get zero weight in the reference -> identical result).

#define N_BATCH  4
#define W_IN     256
#define G_PAD    363
#define N_THETA  180

typedef unsigned int u32;
typedef u32 u32x4 __attribute__((ext_vector_type(4)));
typedef int  i32x4 __attribute__((ext_vector_type(4)));
typedef int  i32x8 __attribute__((ext_vector_type(8)));

typedef __attribute__((address_space(3))) float lds_f32;

__global__ __launch_bounds__(384, 1)
void radon_kernel(const float* __restrict__ x, float* __restrict__ out)
{
    // Whole 256x256 f32 image in LDS: 256 KB (fits in CDNA5's 320 KB/WGP).
    __shared__ float tile[W_IN * W_IN];

    const int a = blockIdx.x;                 // angle index 0..179
    const int n = blockIdx.y;                 // batch index 0..3
    const unsigned tid = threadIdx.x;

    const float* img = x + (size_t)n * (W_IN * W_IN);

#if defined(__has_builtin) && __has_builtin(__builtin_amdgcn_tensor_load_to_lds)
    // --- Tensor Data Mover: DMA the image tile global -> LDS (one wave issues) ---
    if (tid < 32u) {
        unsigned long long ga = (unsigned long long)(uintptr_t)img;
        // D# group 0: count=1 (valid), lds_addr=0, global_addr, type=2 ("image")
        u32x4 g0 = { 1u,
                     0u,
                     (u32)ga,
                     ((u32)((ga >> 32) & 0x01FFFFFFu)) | 0x80000000u };
        // D# group 1:
        //  w0: workgroup_mask=0, data_size=2 (4 bytes)
        //  w1: tensor_dim0[15:0]=256 at bits[63:48]
        //  w2: tensor_dim1[15:0]=256 at bits[95:80]
        //  w3: tile_dim0=256 at bits[127:112]
        //  w4: tile_dim1=256 at bits[143:128], tile_dim2=0
        //  w5: tensor_dim0_stride=256 (elements)
        i32x8 g1 = { 0x00020000,
                     256 << 16,
                     256 << 16,
                     256 << 16,
                     256,
                     256,
                     0,
                     0 };
        // Groups 2/3: higher dims unused (tile_dim2/3/4 = 0)
        i32x4 g2 = { 1, 1, 0, 0 };
        i32x4 g3 = { 0, 0, 0, 0 };
#if __clang_major__ >= 23
        i32x8 g4 = { 0, 0, 0, 0, 0, 0, 0, 0 };
        __builtin_amdgcn_tensor_load_to_lds(g0, g1, g2, g3, g4, 0);
#else
        __builtin_amdgcn_tensor_load_to_lds(g0, g1, g2, g3, 0);
#endif
#if __has_builtin(__builtin_amdgcn_s_wait_tensorcnt)
        __builtin_amdgcn_s_wait_tensorcnt(0);
#else
        asm volatile("s_wait_tensorcnt 0x0" ::: "memory");
#endif
    }
#else
    // Fallback: cooperative vector copy global -> LDS
    for (int i = (int)tid; i < (W_IN * W_IN) / 4; i += (int)blockDim.x) {
        ((float4*)tile)[i] = ((const float4*)img)[i];
    }
#endif
    __syncthreads();

    // Launder the LDS pointer so the compiler cannot reason about the memory
    // behind it (the TDM writes LDS behind clang's back; without this, all
    // tile[] loads fold to undef and the sampling loop is deleted).
    lds_f32* tp = (lds_f32*)tile;
    asm volatile("" : "+v"(tp));

    if (tid < (unsigned)G_PAD) {
        const int w = (int)tid;
        float sn, cs;
        __sincosf((float)a * 0.017453292519943295f, &sn, &cs);

        const float u  = (float)(w - 181);
        const float bx = __builtin_fmaf(cs,  u, 128.0f);   // ix = s*v + bx
        const float by = __builtin_fmaf(-sn, u, 128.0f);   // iy = c*v + by

        float acc = 0.0f;
        #pragma unroll 2
        for (int h = 0; h < G_PAD; ++h) {
            const float v  = (float)(h - 181);
            const float ix = __builtin_fmaf(sn, v, bx);
            const float iy = __builtin_fmaf(cs, v, by);

            const float fx = floorf(ix);
            const float fy = floorf(iy);
            const int   x0 = (int)fx;
            const int   y0 = (int)fy;

            const float wx1 = ix - fx;
            const float wy1 = iy - fy;
            const float wx0 = 1.0f - wx1;
            const float wy0 = 1.0f - wy1;

            const bool inx0 = (unsigned)x0       < 256u;
            const bool inx1 = (unsigned)(x0 + 1) < 256u;
            const bool iny0 = (unsigned)y0       < 256u;
            const bool iny1 = (unsigned)(y0 + 1) < 256u;

            // Clamp addresses so loads are always in-range; mask via cndmask.
            const int r0 = (y0 & 255) << 8;
            const int r1 = ((y0 + 1) & 255) << 8;
            const int c0 = x0 & 255;
            const int c1 = (x0 + 1) & 255;

            const float v00 = (inx0 && iny0) ? tp[r0 + c0] : 0.0f;
            const float v01 = (inx1 && iny0) ? tp[r0 + c1] : 0.0f;
            const float v10 = (inx0 && iny1) ? tp[r1 + c0] : 0.0f;
            const float v11 = (inx1 && iny1) ? tp[r1 + c1] : 0.0f;

            acc = __builtin_fmaf(wy0, __builtin_fmaf(wx0, v00, wx1 * v01),
                  __builtin_fmaf(wy1, __builtin_fmaf(wx0, v10, wx1 * v11), acc));
        }

        // out[n, 0, w, a]
        out[((size_t)n * G_PAD + (size_t)w) * N_THETA + (size_t)a] = acc;
    }
}

extern "C" void kernel_launch(void* const* d_in, const int* in_sizes, int n_in,
                              void* d_out, int out_size, void* d_ws, size_t ws_size,
                              hipStream_t stream) {
    (void)in_sizes; (void)n_in; (void)out_size; (void)d_ws; (void)ws_size;
    const float* x = (const float*)d_in[0];
    float* out = (float*)d_out;

    dim3 grid(N_THETA, N_BATCH, 1);   // one block per (angle, batch)
    dim3 block(384, 1, 1);            // 12 waves; lanes 0..362 own detector bins
    radon_kernel<<<grid, block, 0, stream>>>(x, out);
}